// LinearAttention_4406636445713
// MI455X (gfx1250) — compile-verified
//
#include <hip/hip_runtime.h>
#include <hip/hip_bf16.h>

typedef float v2f __attribute__((ext_vector_type(2)));
typedef float v8f __attribute__((ext_vector_type(8)));
typedef int   v4i __attribute__((ext_vector_type(4)));

#define HEADS 16
#define TLEN  2048
#define DDIM  64
#define MBLK  128          // rows per workgroup (8 waves x 16)
#define NBLK  32           // s-block staged in LDS (double buffered)
#define NSB   (TLEN / NBLK)
#define SQ_ST 68           // LDS row stride (floats) for sQ: conflict-free b64 column reads
#define SV_ST 72           // LDS row stride (floats) for sV: 2*72 % 64 == 16 -> disjoint bank halves
#define SS_ST 20           // LDS row stride for 16-wide S scratch

// Detect the gfx1250 async memory->LDS path (ASYNCcnt-tracked, no VGPR data).
#if defined(__AMDGCN__) && defined(__has_builtin)
#if __has_builtin(__builtin_amdgcn_global_load_async_to_lds_b128)
#define ATTN_ASYNC 1
#endif
#endif
#ifndef ATTN_ASYNC
#define ATTN_ASYNC 0
#endif

// ---------------- Kernel 1: QR = RoPE(Q) ----------------
__global__ __launch_bounds__(256) void rope_qr(const float* __restrict__ Q,
                                               float* __restrict__ QR) {
  int idx = blockIdx.x * 256 + threadIdx.x;          // over 16*2048*64
  int d = idx & (DDIM - 1);
  int t = (idx >> 6) & (TLEN - 1);
  int j = d & 31;
  float inv = __expf(-(float)j * (9.210340371976184f / 32.0f)); // ln(1e4)
  float ang = (float)t * inv;
  float c, s;
  __sincosf(ang, &c, &s);
  float x = Q[idx];
  float y = (d < 32) ? -Q[idx + 32] : Q[idx - 32];
  QR[idx] = x * c + y * s;
}

#if ATTN_ASYNC
typedef __attribute__((address_space(1))) v4i* gp4;   // global int4*
typedef __attribute__((address_space(3))) v4i* lp4;   // LDS    int4*
__device__ __forceinline__ void async_b128(const float* g, float* l) {
  __builtin_amdgcn_global_load_async_to_lds_b128((gp4)g, (lp4)l, 0, 0);
}
__device__ __forceinline__ void wait_async0() {
#if __has_builtin(__builtin_amdgcn_s_wait_asynccnt)
  __builtin_amdgcn_s_wait_asynccnt(0);
#else
  asm volatile("s_wait_asynccnt 0x0" ::: "memory");
#endif
}
#endif

// Per-wave compute on one staged 32-row s-block:
//   GEMM1: S(16x32) = QR_strip(16x64) * QR_s^T   (2 tiles x 16 WMMA)
//   GEMM2: acc(16x64) += S * V_s                  (2 j x 4 jd x 4 k WMMA)
__device__ __forceinline__ void compute_block(const float* __restrict__ sQb,
                                              const float* __restrict__ sVb,
                                              float* __restrict__ mySS,
                                              const v2f aQ[16], v8f acc[4],
                                              int lm, int g) {
#pragma unroll
  for (int j = 0; j < NBLK / 16; ++j) {
    // ---- GEMM1 tile ----
    v8f c = (v8f){};
    const float* bq = sQb + (16 * j + lm) * SQ_ST + 2 * g;     // B[k=d][n=s]
#pragma unroll
    for (int kk = 0; kk < 16; ++kk) {
      v2f b = *reinterpret_cast<const v2f*>(bq + 4 * kk);
      c = __builtin_amdgcn_wmma_f32_16x16x4_f32(
              false, aQ[kk], false, b, (short)0, c, false, false);
    }
    // ---- C-layout -> A-layout via wave-private LDS round trip ----
#pragma unroll
    for (int cv = 0; cv < 8; ++cv)
      mySS[(cv + 8 * g) * SS_ST + lm] = c[cv];                 // S[M][N]
    v2f aS[4];
#pragma unroll
    for (int kk2 = 0; kk2 < 4; ++kk2)
      aS[kk2] = *reinterpret_cast<const v2f*>(&mySS[lm * SS_ST + 4 * kk2 + 2 * g]);
    // ---- GEMM2: B[k=s][n=d] from row-major sV ----
#pragma unroll
    for (int jd = 0; jd < 4; ++jd) {
#pragma unroll
      for (int kk2 = 0; kk2 < 4; ++kk2) {
        int k0 = 16 * j + 4 * kk2 + 2 * g;
        v2f b;
        b.x = sVb[(k0 + 0) * SV_ST + 16 * jd + lm];
        b.y = sVb[(k0 + 1) * SV_ST + 16 * jd + lm];
        acc[jd] = __builtin_amdgcn_wmma_f32_16x16x4_f32(
                      false, aS[kk2], false, b, (short)0, acc[jd], false, false);
      }
    }
  }
}

// ---------------- Kernel 2: out = (QR QR^T) V, streamed over s ----------------
__global__ __launch_bounds__(256) void attn_f32wmma(const float* __restrict__ QR,
                                                    const float* __restrict__ V,
                                                    float* __restrict__ Out) {
  __shared__ float sQ[2][NBLK * SQ_ST];    // 2 x 8.5 KB
  __shared__ float sV[2][NBLK * SV_ST];    // 2 x 9.0 KB (row-major)
  __shared__ float sS[8 * 16 * SS_ST];     // 10 KB per-wave S scratch

  const int tid  = threadIdx.x;
  const int lane = tid & 31;
  const int wv   = tid >> 5;
  const int g    = lane >> 4;
  const int lm   = lane & 15;
  const int h    = blockIdx.y;
  const int mBase = blockIdx.x * MBLK + wv * 16;

  const float* QRh = QR + (size_t)h * TLEN * DDIM;
  const float* Vh  = V  + (size_t)h * TLEN * DDIM;
  float*       Oh  = Out + (size_t)h * TLEN * DDIM;

  // Staging geometry: 2 x b128 per thread per matrix per block
  const int r0 = tid >> 4;                  // f4 = tid      -> row 0..15
  const int r1 = (tid + 256) >> 4;          // f4 = tid+256  -> row 16..31
  const int cc = (tid & 15) << 2;           // col 0..60

  // A-fragments of this wave's 16-row QR strip
  v2f aQ[16];
  {
    const float* rowp = QRh + (size_t)(mBase + lm) * DDIM + 2 * g;
#pragma unroll
    for (int kk = 0; kk < 16; ++kk)
      aQ[kk] = *reinterpret_cast<const v2f*>(rowp + 4 * kk);
  }

  v8f acc[4];
#pragma unroll
  for (int jd = 0; jd < 4; ++jd) acc[jd] = (v8f){};

  float* mySS = sS + wv * 16 * SS_ST;

#if ATTN_ASYNC
  // -------- async double-buffered pipeline: 1 barrier / iteration --------
  {
    async_b128(QRh + (size_t)r0 * DDIM + cc, &sQ[0][r0 * SQ_ST + cc]);
    async_b128(QRh + (size_t)r1 * DDIM + cc, &sQ[0][r1 * SQ_ST + cc]);
    async_b128(Vh  + (size_t)r0 * DDIM + cc, &sV[0][r0 * SV_ST + cc]);
    async_b128(Vh  + (size_t)r1 * DDIM + cc, &sV[0][r1 * SV_ST + cc]);
    wait_async0();
    __syncthreads();
  }
  for (int sb = 0; sb < NSB; ++sb) {
    const int cur = sb & 1, nxt = cur ^ 1;
    if (sb + 1 < NSB) {
      const size_t nb = (size_t)(sb + 1) * NBLK;
      async_b128(QRh + (nb + r0) * DDIM + cc, &sQ[nxt][r0 * SQ_ST + cc]);
      async_b128(QRh + (nb + r1) * DDIM + cc, &sQ[nxt][r1 * SQ_ST + cc]);
      async_b128(Vh  + (nb + r0) * DDIM + cc, &sV[nxt][r0 * SV_ST + cc]);
      async_b128(Vh  + (nb + r1) * DDIM + cc, &sV[nxt][r1 * SV_ST + cc]);
    }
    compute_block(sQ[cur], sV[cur], mySS, aQ, acc, lm, g);
    wait_async0();
    __syncthreads();
  }
#else
  // -------- fallback: register-pipelined prefetch (global loads overlap compute) --------
  float4 rq0, rq1, rv0, rv1;
  {
    rq0 = *reinterpret_cast<const float4*>(QRh + (size_t)r0 * DDIM + cc);
    rq1 = *reinterpret_cast<const float4*>(QRh + (size_t)r1 * DDIM + cc);
    rv0 = *reinterpret_cast<const float4*>(Vh  + (size_t)r0 * DDIM + cc);
    rv1 = *reinterpret_cast<const float4*>(Vh  + (size_t)r1 * DDIM + cc);
    *reinterpret_cast<float4*>(&sQ[0][r0 * SQ_ST + cc]) = rq0;
    *reinterpret_cast<float4*>(&sQ[0][r1 * SQ_ST + cc]) = rq1;
    *reinterpret_cast<float4*>(&sV[0][r0 * SV_ST + cc]) = rv0;
    *reinterpret_cast<float4*>(&sV[0][r1 * SV_ST + cc]) = rv1;
    __syncthreads();
  }
  for (int sb = 0; sb < NSB; ++sb) {
    const int cur = sb & 1, nxt = cur ^ 1;
    if (sb + 1 < NSB) {
      const size_t nb = (size_t)(sb + 1) * NBLK;
      rq0 = *reinterpret_cast<const float4*>(QRh + (nb + r0) * DDIM + cc);
      rq1 = *reinterpret_cast<const float4*>(QRh + (nb + r1) * DDIM + cc);
      rv0 = *reinterpret_cast<const float4*>(Vh  + (nb + r0) * DDIM + cc);
      rv1 = *reinterpret_cast<const float4*>(Vh  + (nb + r1) * DDIM + cc);
    }
    compute_block(sQ[cur], sV[cur], mySS, aQ, acc, lm, g);
    __syncthreads();
    if (sb + 1 < NSB) {
      *reinterpret_cast<float4*>(&sQ[nxt][r0 * SQ_ST + cc]) = rq0;
      *reinterpret_cast<float4*>(&sQ[nxt][r1 * SQ_ST + cc]) = rq1;
      *reinterpret_cast<float4*>(&sV[nxt][r0 * SV_ST + cc]) = rv0;
      *reinterpret_cast<float4*>(&sV[nxt][r1 * SV_ST + cc]) = rv1;
    }
    __syncthreads();
  }
#endif

  // Epilogue: C layout -> global. Row M = cv + 8g, Col N = 16*jd + lm.
#pragma unroll
  for (int jd = 0; jd < 4; ++jd)
#pragma unroll
    for (int cv = 0; cv < 8; ++cv)
      Oh[(size_t)(mBase + cv + 8 * g) * DDIM + 16 * jd + lm] = acc[jd][cv];
}

extern "C" void kernel_launch(void* const* d_in, const int* in_sizes, int n_in,
                              void* d_out, int out_size, void* d_ws, size_t ws_size,
                              hipStream_t stream) {
  const float* Q = (const float*)d_in[0];
  // d_in[1] (K) is unused by the reference: scores = QR @ QR^T
  const float* V = (const float*)d_in[2];
  float* Out = (float*)d_out;
  float* QRws = (float*)d_ws;              // 16*2048*64 floats = 8 MB scratch

  rope_qr<<<(HEADS * TLEN * DDIM) / 256, 256, 0, stream>>>(Q, QRws);

  dim3 grid(TLEN / MBLK, HEADS);           // 16 row-blocks x 16 heads
  attn_f32wmma<<<grid, 256, 0, stream>>>(QRws, V, Out);
}